// ODE_23167053594818
// MI455X (gfx1250) — compile-verified
//
#include <hip/hip_runtime.h>

// Per-sample MLP: out = z + tanh(W5·silu(W4·silu(W3·silu(W2·silu(W1·z+b1)+b2)+b3)+b4)+b5)
// B=256 samples, each with private weights (854080 fp32, ~3.3MB/sample).
// Roofline: 874 MB streamed once vs 436 MFLOP => ~0.5 FLOP/byte, pure HBM-bound
// (23.3 TB/s => ~37us floor). V_WMMA_F32_16X16X4_F32 used as a mat-vec engine.
// Latency hiding: need ~8KB/wave in flight => 8x-unrolled K loop (32 b64 loads
// outstanding) + global_prefetch_b8 one row-length ahead.

namespace {

constexpr int Lc  = 64;
constexpr int Hc  = 512;
constexpr int NLc = 4;
constexpr long long PPS =
    (long long)Lc * Hc + Hc + (long long)(NLc - 1) * (Hc * Hc + Hc) +
    (long long)Hc * Lc + Lc;  // 854080 params per sample

typedef __attribute__((ext_vector_type(2))) float v2f;
typedef __attribute__((ext_vector_type(8))) float v8f;

__device__ __forceinline__ v8f wmma4(v2f a, v2f b, v8f c) {
  // D = A(16x4 f32) * B(4x16 f32) + C(16x16 f32)
  return __builtin_amdgcn_wmma_f32_16x16x4_f32(
      /*neg_a=*/false, a, /*neg_b=*/false, b,
      /*c_mod=*/(short)0, c, /*reuse_a=*/false, /*reuse_b=*/false);
}

__device__ __forceinline__ float silu_f(float x) {
  return x * (1.0f / (1.0f + __expf(-x)));
}

// Wave computes NT 16-row tiles of W[DOUT x DIN] * h (h broadcast into all 16
// columns of the B operand, so every column of D equals the mat-vec result).
// A-layout (32-bit 16x4): lane l -> row (row0 + (l&15)), K-pair 2*(l>>4):
// one aligned b64 load per lane per WMMA. Same float2 pattern for B from LDS.
template <int DIN, int NT>
__device__ __forceinline__ void gemv_tiles(const float* __restrict__ W,
                                           const float* __restrict__ hin,
                                           int row0, int lane, v8f* acc) {
  const int m    = row0 + (lane & 15);
  const int koff = (lane >> 4) << 1;
  const float* w0 = W + (long long)m * DIN + koff;
  const float* hk = hin + koff;
  // Outer step of 32 K-values; inner fully unrolled (8 WMMAs x NT tiles) so the
  // scheduler can keep ~8*NT b64 loads in flight per wave.
  for (int k = 0; k < DIN; k += 32) {
    if (DIN > 64) {
#pragma unroll
      for (int t = 0; t < NT; ++t)  // prefetch one row-length (2KB) ahead
        __builtin_prefetch(w0 + t * 16 * DIN + k + 512, 0, 3);
    }
#pragma unroll
    for (int kk = 0; kk < 32; kk += 4) {
      v2f bv = *(const v2f*)(hk + k + kk);
#pragma unroll
      for (int t = 0; t < NT; ++t) {
        v2f av = *(const v2f*)(w0 + t * 16 * DIN + k + kk);
        acc[t] = wmma4(av, bv, acc[t]);
      }
    }
  }
}

// D layout (16x16 f32 C/D): VGPR r holds row r (lanes 0-15) / row 8+r (16-31),
// all columns identical -> lane 0 emits rows 0..7, lane 16 emits rows 8..15.
__device__ __forceinline__ void emit_silu(const v8f& acc,
                                          const float* __restrict__ bias,
                                          int base0, int lane,
                                          float* __restrict__ hout) {
  if (lane == 0 || lane == 16) {
    const int base = base0 + ((lane == 16) ? 8 : 0);
#pragma unroll
    for (int r = 0; r < 8; ++r)
      hout[base + r] = silu_f(acc[r] + bias[base + r]);
  }
}

}  // namespace

__global__ __launch_bounds__(256, 2) void ODE_persample_mlp(
    const float* __restrict__ params, const float* __restrict__ z,
    float* __restrict__ out, int B) {
  __shared__ float hA[Hc];
  __shared__ float hB[Hc];

  const int b    = blockIdx.x;
  const int tid  = threadIdx.x;
  const int lane = tid & 31;
  const int wave = tid >> 5;  // 8 waves per block
  const float* p = params + (long long)b * PPS;

  if (tid < Lc) hA[tid] = z[b * Lc + tid];
  __syncthreads();

  // ---- layer 1: [H x L], silu ----  (wave owns rows [wave*64, wave*64+64))
  {
    const float* W    = p;
    const float* bias = p + Lc * Hc;
    v8f acc[4] = {};
    const int row0 = wave * 64;
    gemv_tiles<Lc, 4>(W, hA, row0, lane, acc);
#pragma unroll
    for (int t = 0; t < 4; ++t) emit_silu(acc[t], bias, row0 + t * 16, lane, hB);
  }
  __syncthreads();

  // ---- hidden layers: 3 x ([H x H], silu), ping-pong LDS ----
  const float* lp = p + Lc * Hc + Hc;
  float* hin  = hB;
  float* hout = hA;
  for (int li = 0; li < NLc - 1; ++li) {
    const float* W    = lp;
    const float* bias = lp + Hc * Hc;
    v8f acc[4] = {};
    const int row0 = wave * 64;
    gemv_tiles<Hc, 4>(W, hin, row0, lane, acc);
#pragma unroll
    for (int t = 0; t < 4; ++t) emit_silu(acc[t], bias, row0 + t * 16, lane, hout);
    __syncthreads();
    float* tmp = hin; hin = hout; hout = tmp;
    lp += Hc * Hc + Hc;
  }

  // ---- final layer: [L x H], tanh + residual ----  (waves 0..3, one tile each)
  {
    const float* W    = lp;
    const float* bias = lp + Hc * Lc;
    if (wave < 4) {  // wave-uniform: EXEC all-1s inside for WMMA
      v8f acc[1] = {};
      const int row0 = wave * 16;
      gemv_tiles<Hc, 1>(W, hin, row0, lane, acc);
      if (lane == 0 || lane == 16) {
        const int base = row0 + ((lane == 16) ? 8 : 0);
#pragma unroll
        for (int r = 0; r < 8; ++r) {
          out[b * Lc + base + r] =
              z[b * Lc + base + r] + tanhf(acc[0][r] + bias[base + r]);
        }
      }
    }
  }
}

extern "C" void kernel_launch(void* const* d_in, const int* in_sizes, int n_in,
                              void* d_out, int out_size, void* d_ws, size_t ws_size,
                              hipStream_t stream) {
  // inputs (setup_inputs order): t [1], params [B*P], z [B*L]
  const float* params = (const float*)d_in[1];
  const float* z      = (const float*)d_in[2];
  float* out          = (float*)d_out;
  const int B = in_sizes[2] / Lc;  // 256

  dim3 grid(B), block(256);
  hipLaunchKernelGGL(ODE_persample_mlp, grid, block, 0, stream,
                     params, z, out, B);
}